// ACE_layer_83262236000970
// MI455X (gfx1250) — compile-verified
//
#include <hip/hip_runtime.h>

// ACE layer for MI455X (gfx1250), wave32 + V_WMMA_F32_16X16X4_F32.
//
// v_i = sum_{c,a,b,m} w[c,a,b] * ct[a,b,m] * f0[i,c,a,m] * f1[i,c,b,m]
// Per (c,m): M[a,b] = w[c,a,b]*ct[a,b,m];  G[i,a] = sum_b M[a,b] f1[i,c,b,m]
// (16-atom x 16-a WMMA tiles, K=4 steps over b), then acc += f0[i,c,a,m]*G.
//
// Roofline: f0+f1 = 1.68 GB -> ~72us @ 23.3 TB/s; compute (all-4-elements,
// 33 GFLOP) is far below fp32-WMMA ceiling -> memory bound, deterministic.

#define N_ATOMS 8192
#define C_DIM   32
#define NMAX    32
#define LMMAX   25
#define CSPLIT  4                 // split c-reduction over gridDim.y
#define C_PER   (C_DIM / CSPLIT)  // 8 c's per block
#define ATOMS_PER_WAVE  16
#define WAVES_PER_BLOCK 4
#define ATOMS_PER_BLOCK (ATOMS_PER_WAVE * WAVES_PER_BLOCK)  // 64
#define MROW   33                 // 32 + 1 pad -> conflict-reduced LDS rows
#define MPLANE (32 * MROW)        // one m-slice of M (32x32 padded)

typedef __attribute__((ext_vector_type(2))) float v2f;
typedef __attribute__((ext_vector_type(8))) float v8f;

template<int N, int LM>
__device__ __forceinline__ void process_element(
    int e,
    const float* __restrict__ f0, const float* __restrict__ f1,
    const float* __restrict__ w,  const float* __restrict__ ct,
    const int*   __restrict__ eids, float* __restrict__ partial,
    float* Msh, int c0, int atomBase, int lane, int tid)
{
  constexpr int KB = N / 4;          // K-steps per (c,m)
  constexpr int NJ = (N + 15) / 16;  // a-chunks (zero-padded rows cover a>=N)

  v8f acc = {};                       // per-lane partial of sum_a f0*G
  const int lan   = lane & 15;
  const int khalf = (lane >= 16) ? 2 : 0;   // A/B K split by lane half

  for (int c = c0; c < c0 + C_PER; ++c) {
    __syncthreads();
    // Cooperatively build Msh[m][a][b] = w[c,a,b]*ct[a,b,m], padded to 32x32.
    for (int idx = tid; idx < LM * 1024; idx += 128) {
      int m = idx >> 10;
      int ab = idx & 1023;
      int a = ab >> 5, b = ab & 31;
      float v = 0.0f;
      if (a < N && b < N)
        v = w[(c * N + a) * N + b] * ct[(a * N + b) * LM + m];
      Msh[m * MPLANE + a * MROW + b] = v;
    }
    __syncthreads();

    const float* f1b =
        f1 + ((size_t)(atomBase + lan) * C_DIM + c) * (NMAX * LMMAX);

    for (int m = 0; m < LM; ++m) {
      // Stage A-operand (f1) registers: lane%16 = atom row M, K by lane half.
      float A0s[KB], A1s[KB];
#pragma unroll
      for (int kb = 0; kb < KB; ++kb) {
        int k = kb * 4 + khalf;
        A0s[kb] = f1b[k * LMMAX + m];
        A1s[kb] = f1b[(k + 1) * LMMAX + m];
      }
      const float* mplane = Msh + m * MPLANE;

#pragma unroll
      for (int ja = 0; ja < NJ; ++ja) {
        const int a = lan + 16 * ja;            // B-operand N index
        const float* mrow = mplane + a * MROW;
        v8f G = {};
#pragma unroll
        for (int kb = 0; kb < KB; ++kb) {
          int k = kb * 4 + khalf;
          v2f A; A.x = A0s[kb];  A.y = A1s[kb];     // f1[atom, b=k], b=k+1
          v2f B; B.x = mrow[k];  B.y = mrow[k + 1]; // M[a, b=k],   b=k+1
          // D = A(16x4) * B(4x16) + C, fp32
          G = __builtin_amdgcn_wmma_f32_16x16x4_f32(
                  false, A, false, B, (short)0, G, false, false);
        }
        // Contract with f0: D layout -> VGPR r: atom = r (+8 hi half), N=lan.
        const int atomD = atomBase + ((lane >= 16) ? 8 : 0);
#pragma unroll
        for (int r = 0; r < 8; ++r) {
          float f0v =
              f0[(((size_t)(atomD + r) * C_DIM + c) * NMAX + a) * LMMAX + m];
          acc[r] += f0v * G[r];
        }
      }
    }
  }

  // Reduce over the a-dimension: sum across each 16-lane half.
#pragma unroll
  for (int r = 0; r < 8; ++r) {
    float v = acc[r];
    v += __shfl_xor(v, 1, 32);
    v += __shfl_xor(v, 2, 32);
    v += __shfl_xor(v, 4, 32);
    v += __shfl_xor(v, 8, 32);
    acc[r] = v;
  }
  // Lane (half, lan==r) owns atom r (+8 for hi half).
  float vout = 0.0f;
#pragma unroll
  for (int r = 0; r < 8; ++r)
    vout = (lan == r) ? acc[r] : vout;
  if (lan < 8) {
    int atom = atomBase + ((lane >= 16) ? 8 : 0) + lan;
    if (eids[atom] == e) partial[atom] = vout;   // exactly one e matches
  }
}

__global__ void __launch_bounds__(128)
ace_main(const float* __restrict__ f0, const float* __restrict__ f1,
         const float* __restrict__ w0, const float* __restrict__ ct0,
         const float* __restrict__ w1, const float* __restrict__ ct1,
         const float* __restrict__ w2, const float* __restrict__ ct2,
         const float* __restrict__ w3, const float* __restrict__ ct3,
         const int* __restrict__ eids, float* __restrict__ partial)
{
  extern __shared__ float Msh[];   // LMMAX * MPLANE floats = 105,600 B
  const int tid  = threadIdx.x;
  const int lane = tid & 31;
  const int wave = tid >> 5;
  const int atomBase = blockIdx.x * ATOMS_PER_BLOCK + wave * ATOMS_PER_WAVE;
  const int c0 = blockIdx.y * C_PER;
  float* part = partial + (size_t)blockIdx.y * N_ATOMS;

  process_element<32, 25>(0, f0, f1, w0, ct0, eids, part, Msh, c0, atomBase, lane, tid);
  process_element<24, 16>(1, f0, f1, w1, ct1, eids, part, Msh, c0, atomBase, lane, tid);
  process_element<32, 25>(2, f0, f1, w2, ct2, eids, part, Msh, c0, atomBase, lane, tid);
  process_element<16,  9>(3, f0, f1, w3, ct3, eids, part, Msh, c0, atomBase, lane, tid);
}

__global__ void __launch_bounds__(256)
ace_reduce(const float* __restrict__ partial, float* __restrict__ out)
{
  int i = blockIdx.x * blockDim.x + threadIdx.x;
  float s = 0.0f;
#pragma unroll
  for (int p = 0; p < CSPLIT; ++p)
    s += partial[(size_t)p * N_ATOMS + i];
  out[i] = s;   // output shape (N_ATOMS, 1)
}

extern "C" void kernel_launch(void* const* d_in, const int* in_sizes, int n_in,
                              void* d_out, int out_size, void* d_ws, size_t ws_size,
                              hipStream_t stream) {
  // setup_inputs() dict order: f0, f1, w0, ct0, w1, ct1, w2, ct2, w3, ct3, element_ids
  const float* f0  = (const float*)d_in[0];
  const float* f1  = (const float*)d_in[1];
  const float* w0  = (const float*)d_in[2];
  const float* ct0 = (const float*)d_in[3];
  const float* w1  = (const float*)d_in[4];
  const float* ct1 = (const float*)d_in[5];
  const float* w2  = (const float*)d_in[6];
  const float* ct2 = (const float*)d_in[7];
  const float* w3  = (const float*)d_in[8];
  const float* ct3 = (const float*)d_in[9];
  const int*   eids = (const int*)d_in[10];
  float* out     = (float*)d_out;
  float* partial = (float*)d_ws;   // CSPLIT * N_ATOMS floats = 128 KB

  dim3 grid(N_ATOMS / ATOMS_PER_BLOCK, CSPLIT);  // (128, 4)
  size_t shmem = (size_t)LMMAX * MPLANE * sizeof(float);
  hipLaunchKernelGGL(ace_main, grid, dim3(128), shmem, stream,
                     f0, f1, w0, ct0, w1, ct1, w2, ct2, w3, ct3, eids, partial);
  hipLaunchKernelGGL(ace_reduce, dim3(N_ATOMS / 256), dim3(256), 0, stream,
                     partial, out);
}